// MLE_86397562127157
// MI455X (gfx1250) — compile-verified
//
#include <hip/hip_runtime.h>
#include <hip/hip_bf16.h>
#include <math.h>

#define N 512
#define NP1 513
#define NP2 514
#define ALPHA_C 15.0f
#define NEG_C (-1e20f)

typedef __attribute__((ext_vector_type(2))) float v2f;
typedef __attribute__((ext_vector_type(8))) float v8f;

#define PI_AT(pi, i, j, k) pi[(((i) * NP2) + (j)) * 3 + (k)]

// ---------------------------------------------------------------------------
// Kernel 1 (dual): two independent LSE wavefront DPs, one per block, so the
// GT-chain and theta-chain run concurrently on two WGPs.
//   mu[i,j] = ALPHA*W[i-1,j-1] + lse3(mu[i-1,j], mu[i,j-1], mu[i-1,j-1])
// Rolling 3-diagonal buffers live in LDS; block 0 streams full mu(GT) to L2.
// ---------------------------------------------------------------------------
__global__ void lse_dp_dual_kernel(const float* __restrict__ GT,
                                   const float* __restrict__ theta,
                                   float* __restrict__ muGT,
                                   float* __restrict__ corners) {
    __shared__ float diag[3][NP1];
    const int t = threadIdx.x;
    const int isGT = (blockIdx.x == 0);
    const float* W = isGT ? GT : theta;
    float* corner = corners + blockIdx.x;

    for (int d = 0; d <= 2 * N; ++d) {
        float* cur = diag[d % 3];
        float* p1  = diag[(d + 2) % 3];  // diagonal d-1
        float* p2  = diag[(d + 1) % 3];  // diagonal d-2
        const int ilo = (d > N) ? (d - N) : 0;
        const int ihi = (d < N) ? d : N;
        for (int i = ilo + t; i <= ihi; i += blockDim.x) {
            const int j = d - i;
            float v;
            if (i == 0 && j == 0) {
                v = 0.0f;
            } else if (i == 0 || j == 0) {
                v = NEG_C;
            } else {
                const float up   = p1[i - 1];
                const float left = p1[i];
                const float dg   = p2[i - 1];
                float m = fmaxf(fmaxf(up, left), dg);
                v = ALPHA_C * W[(i - 1) * N + (j - 1)] + m +
                    logf(expf(up - m) + expf(left - m) + expf(dg - m));
            }
            cur[i] = v;
            if (isGT) muGT[i * NP1 + j] = v;
            if (i == N && j == N) *corner = v;
        }
        __syncthreads();
    }
}

// ---------------------------------------------------------------------------
// Kernel 2: transition probabilities pi (514 x 514 x 3), zero-padded borders.
// ---------------------------------------------------------------------------
__global__ void pi_kernel(const float* __restrict__ GT,
                          const float* __restrict__ mu,
                          float* __restrict__ pi) {
    const int idx = blockIdx.x * blockDim.x + threadIdx.x;
    if (idx >= NP2 * NP2) return;
    const int i = idx / NP2;
    const int j = idx % NP2;
    float p0 = 0.0f, p1 = 0.0f, p2 = 0.0f;
    if (i >= 1 && i <= N && j >= 1 && j <= N) {
        const float base = ALPHA_C * GT[(i - 1) * N + (j - 1)] - mu[i * NP1 + j];
        p0 = expf(mu[(i - 1) * NP1 + j] + base);       // up parent
        p1 = expf(mu[i * NP1 + (j - 1)] + base);       // left parent
        p2 = expf(mu[(i - 1) * NP1 + (j - 1)] + base); // diag parent
    } else if ((i == 0 && j == 0) || (i == NP2 - 1 && j == NP2 - 1)) {
        p2 = 1.0f;  // e = [0,0,1]
    }
    pi[idx * 3 + 0] = p0;
    pi[idx * 3 + 1] = p1;
    pi[idx * 3 + 2] = p2;
}

// ---------------------------------------------------------------------------
// Kernel 3 (dual): forward DP lam (block 0) and backward DP rho (block 1)
// run concurrently on two WGPs — both depend only on pi.
//   lam[i,j] = up*pi[i,j,0] + left*pi[i,j,1] + diag*pi[i,j,2]
//   rho[i,j] = rho[i+1,j]*pi[i+2,j+1,0] + rho[i,j+1]*pi[i+1,j+2,1]
//            + rho[i+1,j+1]*pi[i+2,j+2,2];  rho[N,N]=1, last row/col else 0.
// ---------------------------------------------------------------------------
__global__ void linear_dp_dual_kernel(const float* __restrict__ pi,
                                      float* __restrict__ lam,
                                      float* __restrict__ rho) {
    __shared__ float diag[3][NP1];
    const int t = threadIdx.x;

    if (blockIdx.x == 0) {
        // ---- forward: lam ----
        for (int d = 0; d <= 2 * N; ++d) {
            float* cur = diag[d % 3];
            float* p1  = diag[(d + 2) % 3];
            float* p2  = diag[(d + 1) % 3];
            const int ilo = (d > N) ? (d - N) : 0;
            const int ihi = (d < N) ? d : N;
            for (int i = ilo + t; i <= ihi; i += blockDim.x) {
                const int j = d - i;
                float v;
                if (i == 0 && j == 0)      v = 1.0f;
                else if (i == 0 || j == 0) v = 0.0f;
                else {
                    v = p1[i - 1] * PI_AT(pi, i, j, 0)
                      + p1[i]     * PI_AT(pi, i, j, 1)
                      + p2[i - 1] * PI_AT(pi, i, j, 2);
                }
                cur[i] = v;
                lam[i * NP1 + j] = v;
            }
            __syncthreads();
        }
    } else {
        // ---- backward: rho ----
        for (int dd = 0; dd <= 2 * N; ++dd) {
            const int d = 2 * N - dd;            // d: 2N .. 0
            float* cur = diag[d % 3];
            float* p1  = diag[(d + 1) % 3];      // diagonal d+1
            float* p2  = diag[(d + 2) % 3];      // diagonal d+2
            const int ilo = (d > N) ? (d - N) : 0;
            const int ihi = (d < N) ? d : N;
            for (int i = ilo + t; i <= ihi; i += blockDim.x) {
                const int j = d - i;
                float v;
                if (i == N && j == N)      v = 1.0f;
                else if (i == N || j == N) v = 0.0f;
                else {
                    v = p1[i + 1] * PI_AT(pi, i + 2, j + 1, 0)
                      + p1[i]     * PI_AT(pi, i + 1, j + 2, 1)
                      + p2[i + 1] * PI_AT(pi, i + 2, j + 2, 2);
                }
                cur[i] = v;
                rho[i * NP1 + j] = v;
            }
            __syncthreads();
        }
    }
}

// ---------------------------------------------------------------------------
// Kernel 4: WMMA reduction of  v(i,j) = osum(i,j) * (GT - theta).
// Each wave packs 64 f32 values per step into a 16x4 A-matrix and multiplies
// by an all-ones 4x16 B with V_WMMA_F32_16X16X4_F32, accumulating in C.
// D[m,n] = rowsum(A)  =>  sum(C over all lanes/regs) = 16 * sum(values).
// 64 blocks x 256 threads = 512 waves, 8 tiles each -> 4096 tiles = 512*512.
// EXEC is all-ones throughout (uniform control flow), as WMMA requires.
// ---------------------------------------------------------------------------
__global__ void reduce_wmma_kernel(const float* __restrict__ theta,
                                   const float* __restrict__ GT,
                                   const float* __restrict__ pi,
                                   const float* __restrict__ lam,
                                   const float* __restrict__ rho,
                                   float* __restrict__ partials) {
    const int tid   = threadIdx.x;
    const int lane  = tid & 31;
    const int wave  = (blockIdx.x * blockDim.x + tid) >> 5;  // 0..511

    v8f c = {};
    v2f ones;
    ones.x = 1.0f; ones.y = 1.0f;

    for (int it = 0; it < 8; ++it) {
        const int tile = wave * 8 + it;           // 0..4095
        const int e0 = tile * 64 + lane * 2;      // two elements per lane
        v2f a;
        #pragma unroll
        for (int q = 0; q < 2; ++q) {
            const int e = e0 + q;
            const int i = e >> 9;                 // / 512
            const int j = e & (N - 1);            // % 512
            const float osum = rho[i * NP1 + j] *
                (lam[i * NP1 + (j + 1)] * PI_AT(pi, i + 1, j + 1, 0) +
                 lam[(i + 1) * NP1 + j] * PI_AT(pi, i + 1, j + 1, 1) +
                 lam[i * NP1 + j]       * PI_AT(pi, i + 1, j + 1, 2));
            const float v = osum * (GT[e] - theta[e]);
            if (q == 0) a.x = v; else a.y = v;
        }
        // D = A x Ones + C   (f32 WMMA, 8 args)
        c = __builtin_amdgcn_wmma_f32_16x16x4_f32(false, a, false, ones,
                                                  (short)0, c, false, false);
    }

    float s = c[0] + c[1] + c[2] + c[3] + c[4] + c[5] + c[6] + c[7];

    __shared__ float sdata[256];
    sdata[tid] = s;
    __syncthreads();
    for (int off = 128; off > 0; off >>= 1) {
        if (tid < off) sdata[tid] += sdata[tid + off];
        __syncthreads();
    }
    if (tid == 0) partials[blockIdx.x] = sdata[0];  // 16x over-counted
}

// ---------------------------------------------------------------------------
// Kernel 5: deterministic in-order finalization.
// ---------------------------------------------------------------------------
__global__ void final_kernel(const float* __restrict__ partials,
                             const float* __restrict__ corners,
                             float* __restrict__ out) {
    if (threadIdx.x == 0 && blockIdx.x == 0) {
        float grand = 0.0f;
        for (int b = 0; b < 64; ++b) grand += partials[b];
        grand *= (1.0f / 16.0f);                 // undo B=ones replication
        // logq - logp = ALPHA*sum((GT-theta)*osum) - muGT[-1,-1] + muTheta[-1,-1]
        out[0] = ALPHA_C * grand - corners[0] + corners[1];
    }
}

extern "C" void kernel_launch(void* const* d_in, const int* in_sizes, int n_in,
                              void* d_out, int out_size, void* d_ws, size_t ws_size,
                              hipStream_t stream) {
    const float* theta = (const float*)d_in[0];
    const float* GT    = (const float*)d_in[1];
    float* out = (float*)d_out;

    float* ws = (float*)d_ws;
    float* muGT     = ws;                                  // 513*513
    float* pi       = muGT + NP1 * NP1;                    // 514*514*3
    float* lam      = pi + NP2 * NP2 * 3;                  // 513*513
    float* rho      = lam + NP1 * NP1;                     // 513*513
    float* corners  = rho + NP1 * NP1;                     // [0]=muGT, [1]=muTheta
    float* partials = corners + 2;                         // 64

    // 1) mu over GT (block 0, stored fully) and mu over theta (block 1),
    //    concurrently on two WGPs.
    lse_dp_dual_kernel<<<2, 512, 0, stream>>>(GT, theta, muGT, corners);
    // 2) transition probabilities
    const int ncells = NP2 * NP2;
    pi_kernel<<<(ncells + 255) / 256, 256, 0, stream>>>(GT, muGT, pi);
    // 3) forward (lam) and backward (rho) marginal DPs, concurrently.
    linear_dp_dual_kernel<<<2, 512, 0, stream>>>(pi, lam, rho);
    // 4) WMMA-based weighted reduction -> 64 block partials
    reduce_wmma_kernel<<<64, 256, 0, stream>>>(theta, GT, pi, lam, rho, partials);
    // 5) deterministic finalize
    final_kernel<<<1, 1, 0, stream>>>(partials, corners, out);
}